// GCNEncoder_6373731467789
// MI455X (gfx1250) — compile-verified
//
#include <hip/hip_runtime.h>
#include <hip/hip_bf16.h>

typedef __attribute__((ext_vector_type(2))) float v2f;
typedef __attribute__((ext_vector_type(8))) float v8f;

#define GCN_N 100000
#define GCN_E 1600000

// ---------------------------------------------------------------- fill zero
__global__ void k_fill0(float* __restrict__ p, long long n) {
  long long i = (long long)blockIdx.x * blockDim.x + threadIdx.x;
  if (i < n) p[i] = 0.0f;
}

// ---------------------------------------------------------------- pos MLP: 64 -> 16 (relu) -> 16
__global__ void k_pe_mlp(const float* __restrict__ pos,
                         const float* __restrict__ w1, const float* __restrict__ b1,
                         const float* __restrict__ w2, const float* __restrict__ b2,
                         float* __restrict__ out, int nrows) {
  int n = blockIdx.x * blockDim.x + threadIdx.x;
  if (n >= nrows) return;
  float t[16];
#pragma unroll
  for (int j = 0; j < 16; ++j) t[j] = b1[j];
  const float* p = pos + (long long)n * 64;
  for (int k = 0; k < 64; ++k) {
    float pv = p[k];
#pragma unroll
    for (int j = 0; j < 16; ++j) t[j] = fmaf(pv, w1[k * 16 + j], t[j]);
  }
#pragma unroll
  for (int j = 0; j < 16; ++j) t[j] = fmaxf(t[j], 0.0f);
  float* o = out + (long long)n * 16;
#pragma unroll
  for (int j = 0; j < 16; ++j) {
    float acc = b2[j];
#pragma unroll
    for (int i = 0; i < 16; ++i) acc = fmaf(t[i], w2[i * 16 + j], acc);
    o[j] = acc;
  }
}

// ---------------------------------------------------------------- degree accumulation over E edges
__global__ void k_deg(const int* __restrict__ row, const float* __restrict__ ew,
                      float* __restrict__ deg, int ne) {
  int e = blockIdx.x * blockDim.x + threadIdx.x;
  if (e >= ne) return;
  __hip_atomic_fetch_add(&deg[row[e]], ew[e], __ATOMIC_RELAXED, __HIP_MEMORY_SCOPE_AGENT);
}

// ---------------------------------------------------------------- dinv = (deg + 1 selfloop)^-0.5
__global__ void k_dinv(const float* __restrict__ deg, float* __restrict__ dinv, int n) {
  int i = blockIdx.x * blockDim.x + threadIdx.x;
  if (i >= n) return;
  float d = deg[i] + 1.0f;          // self-loop weight 1
  d = fmaxf(d, 1e-12f);
  dinv[i] = 1.0f / sqrtf(d);
}

// ---------------------------------------------------------------- WMMA f32 GEMM with LDS-staged A:
// out[N, OUTC] = concat(mainA[N,128], pe[N,16]) @ W[144, OUTC] + bias
// Block = (OUTC/16) waves, one 16-row tile per block; each wave owns a 16-col chunk.
// A tile (16 x 144 f32, 9216 B) staged in LDS via CDNA5 async copy
// (global_load_async_to_lds_b128, ASYNCcnt). B chunk preloaded into VGPRs
// (72 f32/lane). Inner loop: 36 x { ds_load_b64 (A frag) + v_wmma_f32_16x16x4_f32 }.
// Fragment layouts per ISA 7.12.2:
//   A 16x4 f32: lanes 0-15 {K,K+1}, lanes 16-31 {K+2,K+3}, M = lane&15
//   B 4x16 f32: lanes 0-15 rows {K,K+1}, lanes 16-31 rows {K+2,K+3}, N = lane&15
//   C/D 16x16 : VGPR j -> M = j + 8*(lane>=16), N = lane&15
template <int OUTC>
__global__ void k_gemm_wmma(const float* __restrict__ mainA, const float* __restrict__ pe,
                            const float* __restrict__ W, const float* __restrict__ bias,
                            float* __restrict__ out, int nrows) {
  constexpr int K  = 144;
  constexpr int D1 = 128;
  constexpr int KSTEPS = K / 4;                 // 36 WMMAs per tile
  __shared__ __align__(16) float sA[16 * K];    // 9216 bytes

  const int tid  = threadIdx.x;
  const int row0 = blockIdx.x * 16;

  // ---- stage A tile into LDS with async 16B copies (concat resolved here) ----
  for (int t4 = tid; t4 < 16 * (K / 4); t4 += blockDim.x) {
    const int r = t4 / (K / 4);
    const int c = (t4 - r * (K / 4)) * 4;       // float column, multiple of 4
    const int row = row0 + r;
    float* ldsp = &sA[r * K + c];
    const unsigned ldsOff = (unsigned)(unsigned long long)(uintptr_t)ldsp; // flat addr[31:0] == LDS offset
    if (row < nrows) {
      const float* src = (c < D1) ? (mainA + (long long)row * D1 + c)
                                  : (pe   + (long long)row * 16 + (c - D1));
      const unsigned long long ga = (unsigned long long)(uintptr_t)src;
      asm volatile("global_load_async_to_lds_b128 %0, %1, off"
                   :: "v"(ldsOff), "v"(ga) : "memory");
    } else {
      float4 z = make_float4(0.f, 0.f, 0.f, 0.f);
      *reinterpret_cast<float4*>(ldsp) = z;     // disjoint rows -> no race with async writes
    }
  }
  asm volatile("s_wait_asynccnt 0x0" ::: "memory");
  __syncthreads();

  const int lane   = tid & 31;
  const int nchunk = tid >> 5;                  // wave id == 16-col chunk
  const int half   = lane >> 4;
  const int l16    = lane & 15;
  const int col    = nchunk * 16 + l16;

  // ---- preload this wave's B chunk into registers (constant offsets) ----
  const float* Wc = W + col;
  v2f bfrag[KSTEPS];
#pragma unroll
  for (int s = 0; s < KSTEPS; ++s) {
    const int kk = 4 * s + (half << 1);
    bfrag[s].x = Wc[kk * OUTC];
    bfrag[s].y = Wc[(kk + 1) * OUTC];
  }

  // ---- 36 x (LDS A-fragment load + WMMA) ----
  const float* sArow = &sA[l16 * K + (half << 1)];
  v8f acc = {};
#pragma unroll
  for (int s = 0; s < KSTEPS; ++s) {
    v2f a = *reinterpret_cast<const v2f*>(sArow + 4 * s);   // 8B-aligned ds_load_b64
    acc = __builtin_amdgcn_wmma_f32_16x16x4_f32(false, a, false, bfrag[s],
                                                (short)0, acc, false, false);
  }

  // ---- store D + bias ----
  const float bv = bias[col];
#pragma unroll
  for (int j = 0; j < 8; ++j) {
    const int r = row0 + j + (half << 3);
    if (r < nrows) out[(long long)r * OUTC + col] = acc[j] + bv;
  }
}

// ---------------------------------------------------------------- edge aggregation:
// one wave per (edge or self-loop); agg[row] += hh[col] * dinv[row]*w*dinv[col]
__global__ void k_edge_agg(const int* __restrict__ er, const int* __restrict__ ec,
                           const float* __restrict__ ew, const float* __restrict__ dinv,
                           const float* __restrict__ hh, float* __restrict__ agg,
                           int ne, int nn, int C) {
  const int lane = threadIdx.x & 31;
  const int wid  = blockIdx.x * (blockDim.x >> 5) + (threadIdx.x >> 5);
  if (wid >= ne + nn) return;
  int r, c; float w;
  if (wid < ne) { r = er[wid]; c = ec[wid]; w = ew[wid]; }
  else          { r = wid - ne; c = r;      w = 1.0f;    }
  const float norm = dinv[r] * w * dinv[c];
  const int per  = C >> 5;          // 4 (C=128) or 2 (C=64) channels per lane
  const int base = lane * per;
  const float* src = hh  + (long long)c * C + base;
  float*       dst = agg + (long long)r * C + base;
  for (int i = 0; i < per; ++i) {
    __hip_atomic_fetch_add(&dst[i], src[i] * norm, __ATOMIC_RELAXED, __HIP_MEMORY_SCOPE_AGENT);
  }
}

// ---------------------------------------------------------------- fused LayerNorm(128) + ReLU, in place
__global__ void k_ln_relu(float* __restrict__ h, const float* __restrict__ g,
                          const float* __restrict__ b, int nrows) {
  const int lane = threadIdx.x & 31;
  const int wid  = blockIdx.x * (blockDim.x >> 5) + (threadIdx.x >> 5);
  if (wid >= nrows) return;
  float* rowp = h + (long long)wid * 128;
  float v[4];
  float s = 0.0f;
#pragma unroll
  for (int i = 0; i < 4; ++i) { v[i] = rowp[lane * 4 + i]; s += v[i]; }
#pragma unroll
  for (int m = 16; m > 0; m >>= 1) s += __shfl_xor(s, m, 32);
  const float mu = s * (1.0f / 128.0f);
  float q = 0.0f;
#pragma unroll
  for (int i = 0; i < 4; ++i) { float d = v[i] - mu; q += d * d; }
#pragma unroll
  for (int m = 16; m > 0; m >>= 1) q += __shfl_xor(q, m, 32);
  const float rstd = 1.0f / sqrtf(q * (1.0f / 128.0f) + 1e-5f);
#pragma unroll
  for (int i = 0; i < 4; ++i) {
    const int cdx = lane * 4 + i;
    float o = (v[i] - mu) * rstd * g[cdx] + b[cdx];
    rowp[cdx] = fmaxf(o, 0.0f);
  }
}

// ---------------------------------------------------------------- launch
extern "C" void kernel_launch(void* const* d_in, const int* in_sizes, int n_in,
                              void* d_out, int out_size, void* d_ws, size_t ws_size,
                              hipStream_t stream) {
  const float* x      = (const float*)d_in[0];
  const float* pos    = (const float*)d_in[1];
  const int*   ei     = (const int*)  d_in[2];   // [2, E] flat: rows then cols
  const float* ew     = (const float*)d_in[3];
  const float* pp0w1  = (const float*)d_in[4];
  const float* pp0b1  = (const float*)d_in[5];
  const float* pp0w2  = (const float*)d_in[6];
  const float* pp0b2  = (const float*)d_in[7];
  const float* pp1w1  = (const float*)d_in[8];
  const float* pp1b1  = (const float*)d_in[9];
  const float* pp1w2  = (const float*)d_in[10];
  const float* pp1b2  = (const float*)d_in[11];
  const float* W0     = (const float*)d_in[12];  // [144, 128]
  const float* b0     = (const float*)d_in[13];
  const float* W1     = (const float*)d_in[14];  // [144, 64]
  const float* b1     = (const float*)d_in[15];
  const float* ln_g   = (const float*)d_in[16];
  const float* ln_b   = (const float*)d_in[17];
  float* out = (float*)d_out;

  const int N = GCN_N, E = GCN_E;
  float* ws   = (float*)d_ws;
  float* pe0  = ws;                             // N*16
  float* pe1  = pe0  + (size_t)N * 16;          // N*16
  float* deg  = pe1  + (size_t)N * 16;          // N
  float* dinv = deg  + (size_t)N;               // N
  float* hh0  = dinv + (size_t)N;               // N*128 (reused as hh1)
  float* agg0 = hh0  + (size_t)N * 128;         // N*128
  // total: 290*N floats = ~116 MB

  const int T = 256;
  const long long nAgg0 = (long long)N * 128;
  const long long nOut  = (long long)N * 64;

  k_fill0<<<(int)((N + T - 1) / T),       T, 0, stream>>>(deg, N);
  k_fill0<<<(int)((nAgg0 + T - 1) / T),   T, 0, stream>>>(agg0, nAgg0);
  k_fill0<<<(int)((nOut + T - 1) / T),    T, 0, stream>>>(out, nOut);

  k_pe_mlp<<<(N + T - 1) / T, T, 0, stream>>>(pos, pp0w1, pp0b1, pp0w2, pp0b2, pe0, N);
  k_pe_mlp<<<(N + T - 1) / T, T, 0, stream>>>(pos, pp1w1, pp1b1, pp1w2, pp1b2, pe1, N);

  k_deg <<<(E + T - 1) / T, T, 0, stream>>>(ei, ew, deg, E);
  k_dinv<<<(N + T - 1) / T, T, 0, stream>>>(deg, dinv, N);

  const int mtiles = (N + 15) / 16;
  const int wavesPerBlock = T / 32;

  // layer 0: hh0 = concat(x, pe0) @ W0 + b0   (K=144, OUTC=128) -> 8 waves/block
  k_gemm_wmma<128><<<mtiles, 256, 0, stream>>>(x, pe0, W0, b0, hh0, N);
  {
    const int totEdges = E + N;
    k_edge_agg<<<(totEdges + wavesPerBlock - 1) / wavesPerBlock, T, 0, stream>>>(
        ei, ei + E, ew, dinv, hh0, agg0, E, N, 128);
  }
  k_ln_relu<<<(N + wavesPerBlock - 1) / wavesPerBlock, T, 0, stream>>>(agg0, ln_g, ln_b, N);

  // layer 1: hh1 (reuse hh0) = concat(h, pe1) @ W1 + b1   (K=144, OUTC=64) -> 4 waves/block
  k_gemm_wmma<64><<<mtiles, 128, 0, stream>>>(agg0, pe1, W1, b1, hh0, N);
  {
    const int totEdges = E + N;
    k_edge_agg<<<(totEdges + wavesPerBlock - 1) / wavesPerBlock, T, 0, stream>>>(
        ei, ei + E, ew, dinv, hh0, out, E, N, 64);
  }
}